// AngularBasis_84851373899904
// MI455X (gfx1250) — compile-verified
//
#include <hip/hip_runtime.h>
#include <stdint.h>

// Angular basis expansion, L_MAX = 6.
// Per edge: 1093 floats = levels of size 1,3,9,27,81,243,729 at
// offsets 0,1,4,13,40,121,364.
//
// Roofline: 874 MB stores @ 23.3 TB/s => ~37.5us floor; pure store-BW kernel.
// Strategy: build 8-edge tiles (34,976 B, a multiple of 16 B) in LDS, then
// drain LDS -> HBM with CDNA5 async b128 stores (ASYNCcnt path).

#define EDGES_PER_BLOCK 8
#define ROW_LEN 1093
#define TILE_FLOATS (EDGES_PER_BLOCK * ROW_LEN) /* 8744 */
#define TILE_VEC4 (TILE_FLOATS / 4)             /* 2186 */
#define BLOCK_THREADS 256

#if defined(__has_builtin)
#if __has_builtin(__builtin_amdgcn_global_store_async_from_lds_b128)
#define HAVE_ASYNC_STORE 1
#endif
#if __has_builtin(__builtin_amdgcn_s_wait_asynccnt)
#define HAVE_WAIT_ASYNC 1
#endif
#endif

typedef int v4i __attribute__((ext_vector_type(4)));
typedef v4i __attribute__((address_space(1))) * as1_v4i_ptr;  // global int4*
typedef v4i __attribute__((address_space(3))) * as3_v4i_ptr;  // LDS int4*

__global__ __launch_bounds__(BLOCK_THREADS) void angular_basis_kernel(
    const float* __restrict__ dr, float* __restrict__ out, int n) {
  __shared__ __align__(16) float lds[TILE_FLOATS];

  const int tid  = threadIdx.x;
  const int lane = tid & 31;
  const int wave = tid >> 5;
  const int edge = blockIdx.x * EDGES_PER_BLOCK + wave;

  // ---- Phase 1: one wave per edge; lane L (< 27) owns level-3 entry L and
  // its entire subtree. Value of entry k at level l = prod over base-3 digits
  // of k of dr[digit]; products are commutative so digit order is free.
  if (edge < n) {
    const float x = dr[3 * edge + 0];
    const float y = dr[3 * edge + 1];
    const float z = dr[3 * edge + 2];
    float* __restrict__ L = &lds[wave * ROW_LEN];
    if (lane < 27) {
      const int d0 = lane % 3;
      const int d1 = (lane / 3) % 3;
      const int d2 = lane / 9;
      const float f0 = (d0 == 0) ? x : ((d0 == 1) ? y : z);
      const float f1 = (d1 == 0) ? x : ((d1 == 1) ? y : z);
      const float f2 = (d2 == 0) ? x : ((d2 == 1) ? y : z);

      if (lane == 0) L[0] = 1.0f;            // level 0
      if (lane < 3)  L[1 + lane] = f0;       // level 1: dr[lane]
      if (lane < 9)  L[4 + lane] = f1 * f0;  // level 2: dr[k/3]*dr[k%3]
      const float v3 = f2 * f1 * f0;         // level 3
      L[13 + lane] = v3;

      const float d[3] = {x, y, z};
#pragma unroll
      for (int j = 0; j < 3; ++j) {
        const float v4 = v3 * d[j];
        L[40 + 3 * lane + j] = v4;           // level 4: k4 = 3*lane + j
#pragma unroll
        for (int j2 = 0; j2 < 3; ++j2) {
          const float v5 = v4 * d[j2];
          L[121 + 9 * lane + 3 * j + j2] = v5;  // level 5: k5 = 3*k4 + j2
#pragma unroll
          for (int j3 = 0; j3 < 3; ++j3) {
            L[364 + 27 * lane + 9 * j + 3 * j2 + j3] = v5 * d[j3];  // level 6
          }
        }
      }
    }
  }
  __syncthreads();

  // ---- Phase 2: flat copy of the tile LDS -> global. Tile byte size
  // (34,976) and tile base offset are multiples of 16, so b128 works.
  const long long row0 = (long long)blockIdx.x * EDGES_PER_BLOCK;
  const int rows = min(EDGES_PER_BLOCK, n - (int)row0);
  float* __restrict__ gbase = out + row0 * (long long)ROW_LEN;

  if (rows == EDGES_PER_BLOCK) {
#if defined(HAVE_ASYNC_STORE)
    for (int i = tid; i < TILE_VEC4; i += BLOCK_THREADS) {
      __builtin_amdgcn_global_store_async_from_lds_b128(
          (as1_v4i_ptr)(gbase + 4 * i),
          (as3_v4i_ptr)(&lds[4 * i]),
          /*offset=*/0, /*cpol=*/0);
    }
#if defined(HAVE_WAIT_ASYNC)
    __builtin_amdgcn_s_wait_asynccnt(0);
#endif
#else
    const float4* __restrict__ lv = (const float4*)lds;
    float4* __restrict__ gv = (float4*)gbase;
    for (int i = tid; i < TILE_VEC4; i += BLOCK_THREADS) {
      gv[i] = lv[i];
    }
#endif
  } else if (rows > 0) {
    const int cnt = rows * ROW_LEN;
    for (int i = tid; i < cnt; i += BLOCK_THREADS) {
      gbase[i] = lds[i];
    }
  }
}

extern "C" void kernel_launch(void* const* d_in, const int* in_sizes, int n_in,
                              void* d_out, int out_size, void* d_ws, size_t ws_size,
                              hipStream_t stream) {
  const float* dr = (const float*)d_in[0];
  float* out = (float*)d_out;
  const int n = in_sizes[0] / 3;  // dr is [n,3] float32
  if (n <= 0) return;
  const int grid = (n + EDGES_PER_BLOCK - 1) / EDGES_PER_BLOCK;
  angular_basis_kernel<<<grid, BLOCK_THREADS, 0, stream>>>(dr, out, n);
}